// NativelySparseBallAttention_33870112096810
// MI455X (gfx1250) — compile-verified
//
#include <hip/hip_runtime.h>
#include <math.h>

typedef __attribute__((ext_vector_type(2))) float v2f;
typedef __attribute__((ext_vector_type(8))) float v8f;
typedef __attribute__((ext_vector_type(4))) unsigned int v4u;
typedef __attribute__((ext_vector_type(4))) int v4i;
typedef __attribute__((ext_vector_type(8))) int v8i;

#define NMP   8192   // total points
#define DMODEL 256
#define NBALL 256
#define BALLM 32
#define HEADS 8
#define EDIM  32
#define TOPK  4

#if defined(__has_builtin)
#if __has_builtin(__builtin_amdgcn_tensor_load_to_lds)
#define HAVE_TDM 1
#endif
#endif

// ---------------------------------------------------------------------------
// TDM: DMA a 2D tile (tile_d1 rows x tile_d0 f32 elems, row stride
// stride0_elems) from global memory into LDS. Issued by one wave; caller must
// __syncthreads() afterwards. D# packing per CDNA5 ISA 8.3/8.4:
//   G0: [1:0]=count=1, [63:32]=lds_addr, [120:64]=global_addr, [127:126]=type=2
//   G1: [17:16]=data_size(2 -> 4B), [79:48]=tensor_dim0, [111:80]=tensor_dim1,
//       [127:112]=tile_dim0, [143:128]=tile_dim1, [207:160]=dim0_stride
// ---------------------------------------------------------------------------
#ifdef HAVE_TDM
__device__ __forceinline__ void tdm_load_2d(void* lds_ptr, const void* gptr,
                                            unsigned tile_d0, unsigned tile_d1,
                                            unsigned long long stride0_elems) {
    const unsigned lds_off = (unsigned)(unsigned long long)(uintptr_t)lds_ptr;
    const unsigned long long ga = (unsigned long long)(uintptr_t)gptr;
    v4u g0;
    g0[0] = 1u;                                                // count=1, user D#
    g0[1] = lds_off;                                           // lds_addr
    g0[2] = (unsigned)(ga & 0xFFFFFFFFull);                    // global_addr lo
    g0[3] = (unsigned)((ga >> 32) & 0x1FFFFFFull) | (2u << 30);// addr hi | type=2
    v8i g1;
    g1[0] = (int)(2u << 16);                                   // data_size = 4B
    g1[1] = (int)((tile_d0 & 0xFFFFu) << 16);                  // tensor_dim0 lo
    g1[2] = (int)((tile_d0 >> 16) | ((tile_d1 & 0xFFFFu) << 16)); // dim0 hi|dim1 lo
    g1[3] = (int)((tile_d1 >> 16) | (tile_d0 << 16));          // dim1 hi | tile_dim0
    g1[4] = (int)(tile_d1 & 0xFFFFu);                          // tile_dim1, tile_dim2=0
    g1[5] = (int)(unsigned)(stride0_elems & 0xFFFFFFFFull);    // dim0_stride lo
    g1[6] = (int)(unsigned)((stride0_elems >> 32) & 0xFFFFull);// dim0_stride hi
    g1[7] = 0;
    v4i gz = {0, 0, 0, 0};
#if __clang_major__ >= 23
    v8i gz8 = {0, 0, 0, 0, 0, 0, 0, 0};
    __builtin_amdgcn_tensor_load_to_lds(g0, g1, gz, gz, gz8, 0);
#else
    __builtin_amdgcn_tensor_load_to_lds(g0, g1, gz, gz, 0);
#endif
    __builtin_amdgcn_s_wait_tensorcnt(0);
}
#endif

// ---------------------------------------------------------------------------
// 1. positional encoding: x' = x + (pos - ball_mean(pos)) @ W_pe^T + b_pe
// ---------------------------------------------------------------------------
__global__ void pe_kernel(const float* __restrict__ x, const float* __restrict__ pos,
                          const float* __restrict__ W_pe, const float* __restrict__ b_pe,
                          float* __restrict__ XP) {
    __shared__ float s_pos[BALLM * 3];
    __shared__ float s_mean[3];
    const int ball = blockIdx.x;
    const int tid = threadIdx.x;
    if (tid < BALLM * 3) s_pos[tid] = pos[ball * BALLM * 3 + tid];
    __syncthreads();
    if (tid < 3) {
        float s = 0.f;
        #pragma unroll
        for (int r = 0; r < BALLM; ++r) s += s_pos[r * 3 + tid];
        s_mean[tid] = s * (1.f / 32.f);
    }
    __syncthreads();
    for (int i = tid; i < BALLM * DMODEL; i += blockDim.x) {
        const int r = i >> 8;
        const int c = i & 255;
        const float r0 = s_pos[r * 3 + 0] - s_mean[0];
        const float r1 = s_pos[r * 3 + 1] - s_mean[1];
        const float r2 = s_pos[r * 3 + 2] - s_mean[2];
        const int row = ball * BALLM + r;
        XP[(size_t)row * DMODEL + c] =
            x[(size_t)row * DMODEL + c]
            + r0 * W_pe[c * 3 + 0] + r1 * W_pe[c * 3 + 1] + r2 * W_pe[c * 3 + 2]
            + b_pe[c];
    }
}

// ---------------------------------------------------------------------------
// 2. per-ball mean of x'  (k-center computed from mean(x) since k is linear)
// ---------------------------------------------------------------------------
__global__ void xbar_kernel(const float* __restrict__ XP, float* __restrict__ XBAR) {
    const int ball = blockIdx.x;
    const int c = threadIdx.x;
    float s = 0.f;
    #pragma unroll 8
    for (int r = 0; r < BALLM; ++r)
        s += XP[((size_t)ball * BALLM + r) * DMODEL + c];
    XBAR[(size_t)ball * DMODEL + c] = s * (1.f / 32.f);
}

// ---------------------------------------------------------------------------
// 3/4/7. fp32 WMMA GEMM: out = A(Mx256) @ Wsel^T + bias
//   mode 0 (Q):    W row = 3*col     (96h+3e)    out[h][row][e]
//   mode 1 (KC):   W row = 3*col + 1 (96h+3e+1)  out[h][ball][e]
//   mode 2 (PROJ): W row = col                   out[row][col]
// Block = 8 waves, macro-tile 256(M) x 64(N). The 64 weight rows (constant
// row stride 3 or 1) are TDM-DMA'd into LDS once; each wave computes a
// 32x64 register tile (2x4 WMMA accumulators) -> per k-step:
// 2 global_load_b64 + 4 ds_load_b64 feed 8 v_wmma_f32_16x16x4_f32.
// ---------------------------------------------------------------------------
__global__ void wmma_gemm_kernel(const float* __restrict__ A,
                                 const float* __restrict__ W,
                                 const float* __restrict__ bias,
                                 float* __restrict__ out,
                                 int M, int mode) {
    __shared__ float s_b[64 * DMODEL];   // 64 KB weight tile
    const int wave = threadIdx.x >> 5;
    const int lane = threadIdx.x & 31;
    const int hi   = lane >> 4;          // K offset 0/2, M offset 0/8
    const int l16  = lane & 15;

    const int nblk = blockIdx.x & 3;     // 4 N-blocks of 64 over DMODEL=256
    const int mblk = blockIdx.x >> 2;
    const int n0 = nblk * 64;
    const int m_base = mblk * 256 + wave * 32;

    const int rstep = (mode == 2) ? 1 : 3;
    const int wrow0 = (mode == 2) ? n0 : (3 * n0 + mode); // mode1 -> +1 (k rows)

#ifdef HAVE_TDM
    if (wave == 0)
        tdm_load_2d(s_b, W + (size_t)wrow0 * DMODEL,
                    DMODEL, 64, (unsigned long long)rstep * DMODEL);
#else
    for (int i = threadIdx.x; i < 64 * DMODEL; i += blockDim.x) {
        const int r = i >> 8, c = i & 255;
        s_b[i] = W[(size_t)(wrow0 + r * rstep) * DMODEL + c];
    }
#endif
    __syncthreads();

    const float* arow0 = A + (size_t)(m_base + l16) * DMODEL + 2 * hi;
    const float* arow1 = arow0 + (size_t)16 * DMODEL;
    const float* brow = s_b + (size_t)l16 * DMODEL + 2 * hi;  // +16j*DMODEL per N-tile

    v8f acc[2][4];
    #pragma unroll
    for (int i = 0; i < 2; ++i)
        #pragma unroll
        for (int j = 0; j < 4; ++j)
            acc[i][j] = (v8f){0.f, 0.f, 0.f, 0.f, 0.f, 0.f, 0.f, 0.f};

    #pragma unroll 8
    for (int kk = 0; kk < DMODEL; kk += 4) {
        v2f a0, a1, b[4];
        a0[0] = arow0[kk]; a0[1] = arow0[kk + 1];
        a1[0] = arow1[kk]; a1[1] = arow1[kk + 1];
        #pragma unroll
        for (int j = 0; j < 4; ++j) {
            b[j][0] = brow[j * 16 * DMODEL + kk];
            b[j][1] = brow[j * 16 * DMODEL + kk + 1];
        }
        #pragma unroll
        for (int j = 0; j < 4; ++j) {
            acc[0][j] = __builtin_amdgcn_wmma_f32_16x16x4_f32(false, a0, false, b[j],
                                                              (short)0, acc[0][j], false, false);
            acc[1][j] = __builtin_amdgcn_wmma_f32_16x16x4_f32(false, a1, false, b[j],
                                                              (short)0, acc[1][j], false, false);
        }
    }

    #pragma unroll
    for (int i = 0; i < 2; ++i) {
        #pragma unroll
        for (int j = 0; j < 4; ++j) {
            const int col = n0 + 16 * j + l16;
            const float bv = (mode == 0) ? bias[3 * col]
                           : (mode == 1) ? bias[3 * col + 1]
                                         : bias[col];
            #pragma unroll
            for (int v = 0; v < 8; ++v) {
                const int row = m_base + 16 * i + v + 8 * hi;
                const float val = acc[i][j][v] + bv;
                size_t off;
                if (mode == 0)
                    off = (size_t)(col >> 5) * (NMP * EDIM) + (size_t)row * EDIM + (col & 31);
                else if (mode == 1)
                    off = (size_t)(col >> 5) * (NBALL * EDIM) + (size_t)row * EDIM + (col & 31);
                else
                    off = (size_t)row * DMODEL + col;
                out[off] = val;
            }
        }
    }
}

// ---------------------------------------------------------------------------
// 5. sim = Q[h] @ KC[h]^T (WMMA, 16 queries x 256 balls per block) + top-4.
// KC[h] (256x32 = 32 KB, contiguous) is TDM-staged into LDS.
// ---------------------------------------------------------------------------
__global__ void sim_topk_kernel(const float* __restrict__ Q,
                                const float* __restrict__ KC,
                                int* __restrict__ IDX) {
    __shared__ float s_kc[NBALL * EDIM];  // 32 KB
    __shared__ float s_sim[16 * NBALL];   // 16 KB
    const int wave = threadIdx.x >> 5;
    const int lane = threadIdx.x & 31;
    const int hi = lane >> 4, l16 = lane & 15;
    const int h  = blockIdx.x >> 9;       // 512 query-tiles per head
    const int q0 = (blockIdx.x & 511) * 16;

    const float* Qb = Q  + (size_t)h * NMP   * EDIM;
    const float* Kb = KC + (size_t)h * NBALL * EDIM;

#ifdef HAVE_TDM
    if (wave == 0)
        tdm_load_2d(s_kc, Kb, NBALL * EDIM, 1, NBALL * EDIM);
#else
    for (int i = threadIdx.x; i < NBALL * EDIM; i += blockDim.x) s_kc[i] = Kb[i];
#endif
    __syncthreads();

    const float* arow = Qb + (size_t)(q0 + l16) * EDIM + 2 * hi;
    #pragma unroll
    for (int nt = 0; nt < 2; ++nt) {      // 8 waves x 2 = 16 ball-tiles = 256 balls
        const int n0 = (wave * 2 + nt) * 16;
        const float* brow = s_kc + (size_t)(n0 + l16) * EDIM + 2 * hi;
        v8f acc = {0.f, 0.f, 0.f, 0.f, 0.f, 0.f, 0.f, 0.f};
        #pragma unroll
        for (int kk = 0; kk < EDIM; kk += 4) {
            v2f a, b;
            a[0] = arow[kk]; a[1] = arow[kk + 1];
            b[0] = brow[kk]; b[1] = brow[kk + 1];
            acc = __builtin_amdgcn_wmma_f32_16x16x4_f32(false, a, false, b,
                                                        (short)0, acc, false, false);
        }
        #pragma unroll
        for (int v = 0; v < 8; ++v)
            s_sim[(v + 8 * hi) * NBALL + (n0 + l16)] = acc[v];
    }
    __syncthreads();

    if (threadIdx.x < 16) {
        const float* row = s_sim + threadIdx.x * NBALL;
        float tv[TOPK] = {-INFINITY, -INFINITY, -INFINITY, -INFINITY};
        int   ti[TOPK] = {0, 0, 0, 0};
        for (int b = 0; b < NBALL; ++b) {
            const float v = row[b];
            if (v > tv[TOPK - 1]) {                // strict >: jax tie-break (lower idx)
                int j = TOPK - 1;
                while (j > 0 && v > tv[j - 1]) {
                    tv[j] = tv[j - 1]; ti[j] = ti[j - 1]; --j;
                }
                tv[j] = v; ti[j] = b;
            }
        }
        int* o = IDX + ((size_t)h * NMP + q0 + threadIdx.x) * TOPK;
        #pragma unroll
        for (int t = 0; t < TOPK; ++t) o[t] = ti[t];
    }
}

// ---------------------------------------------------------------------------
// 6. gathered attention: one wave per (head, query); k = v = q (per reference)
// ---------------------------------------------------------------------------
__global__ void attn_kernel(const float* __restrict__ Q,
                            const int* __restrict__ IDX,
                            float* __restrict__ ATT) {
    __shared__ float s_q[8][EDIM];
    __shared__ float s_w[8][TOPK * BALLM];
    const int wave = threadIdx.x >> 5;
    const int lane = threadIdx.x & 31;
    const int pair = blockIdx.x * 8 + wave;  // 65536 (h,p) pairs
    const int h = pair >> 13;
    const int p = pair & (NMP - 1);

    const float* Qh = Q + (size_t)h * NMP * EDIM;
    s_q[wave][lane] = Qh[(size_t)p * EDIM + lane];
    __syncthreads();

    const int* idx = IDX + ((size_t)h * NMP + p) * TOPK;
    int balls[TOPK];
    float sc[TOPK];
    const float scale = 0.17677669529663687f;   // 1/sqrt(32)
    #pragma unroll
    for (int t = 0; t < TOPK; ++t) {
        balls[t] = idx[t];
        const float* krow = Qh + ((size_t)balls[t] * BALLM + lane) * EDIM;  // lane = key row
        float s = 0.f;
        #pragma unroll
        for (int e = 0; e < EDIM; ++e) s += krow[e] * s_q[wave][e];
        sc[t] = s * scale;
    }

    // softmax over 128 scores (4 per lane x 32 lanes), wave32 reductions
    float mx = fmaxf(fmaxf(sc[0], sc[1]), fmaxf(sc[2], sc[3]));
    #pragma unroll
    for (int off = 16; off > 0; off >>= 1) mx = fmaxf(mx, __shfl_xor(mx, off, 32));
    float pexp[TOPK];
    float ssum = 0.f;
    #pragma unroll
    for (int t = 0; t < TOPK; ++t) { pexp[t] = expf(sc[t] - mx); ssum += pexp[t]; }
    #pragma unroll
    for (int off = 16; off > 0; off >>= 1) ssum += __shfl_xor(ssum, off, 32);
    const float inv = 1.f / ssum;
    #pragma unroll
    for (int t = 0; t < TOPK; ++t) s_w[wave][t * BALLM + lane] = pexp[t] * inv;
    __syncthreads();

    float acc = 0.f;
    #pragma unroll
    for (int t = 0; t < TOPK; ++t) {
        const float* kbase = Qh + (size_t)balls[t] * BALLM * EDIM;
        #pragma unroll 8
        for (int r = 0; r < BALLM; ++r)
            acc += s_w[wave][t * BALLM + r] * kbase[r * EDIM + lane];  // lane = e
    }
    ATT[(size_t)p * DMODEL + h * EDIM + lane] = acc;
}

// ---------------------------------------------------------------------------
extern "C" void kernel_launch(void* const* d_in, const int* in_sizes, int n_in,
                              void* d_out, int out_size, void* d_ws, size_t ws_size,
                              hipStream_t stream) {
    const float* x      = (const float*)d_in[0];
    const float* pos    = (const float*)d_in[1];
    const float* W_pe   = (const float*)d_in[2];
    const float* b_pe   = (const float*)d_in[3];
    const float* W_qkv  = (const float*)d_in[4];
    const float* b_qkv  = (const float*)d_in[5];
    const float* W_proj = (const float*)d_in[6];
    const float* b_proj = (const float*)d_in[7];
    float* out = (float*)d_out;

    float* XP   = (float*)d_ws;                     // 8192*256
    float* XBAR = XP   + (size_t)NMP * DMODEL;      // 256*256
    float* Qb   = XBAR + (size_t)NBALL * DMODEL;    // 8*8192*32
    float* KC   = Qb   + (size_t)HEADS * NMP * EDIM;    // 8*256*32
    float* ATT  = KC   + (size_t)HEADS * NBALL * EDIM;  // 8192*256
    int*   IDX  = (int*)(ATT + (size_t)NMP * DMODEL);   // 8*8192*4 ints

    pe_kernel<<<NBALL, 256, 0, stream>>>(x, pos, W_pe, b_pe, XP);
    xbar_kernel<<<NBALL, 256, 0, stream>>>(XP, XBAR);
    // Q: M=8192 -> 32 M-blocks x 4 N-blocks
    wmma_gemm_kernel<<<128, 256, 0, stream>>>(XP, W_qkv, b_qkv, Qb, NMP, 0);
    // KC: M=256 -> 1 M-block x 4 N-blocks
    wmma_gemm_kernel<<<4, 256, 0, stream>>>(XBAR, W_qkv, b_qkv, KC, NBALL, 1);
    // sim + top4: 8 heads * 512 query-tiles
    sim_topk_kernel<<<HEADS * (NMP / 16), 256, 0, stream>>>(Qb, KC, IDX);
    // attention: 65536 (h,q) waves / 8 per block
    attn_kernel<<<HEADS * NMP / 8, 256, 0, stream>>>(Qb, IDX, ATT);
    // projection
    wmma_gemm_kernel<<<128, 256, 0, stream>>>(ATT, W_proj, b_proj, out, NMP, 2);
}